// ProbAttention_5196910428224
// MI455X (gfx1250) — compile-verified
//
#include <hip/hip_runtime.h>

typedef __attribute__((ext_vector_type(2))) float v2f;
typedef __attribute__((ext_vector_type(8))) float v8f;

#define B_ 16
#define L_ 4096
#define H_ 8
#define D_ 64
#define SK_ 45      // sample_k
#define NTOP_ 45    // n_top
#define MTILES_ 3   // ceil(45/16)
#define SCALE_ 0.125f
#define NEG_INF_ (-1.0e30f)
#define PART_STRIDE_ 1056  // 16*64 O + 16 m + 16 l
#define PSTRIDE_ 68        // padded LDS row stride for 16x64 P tile

// ---------------------------------------------------------------------------
// Kernel 1a: vmean[b,h,d] = mean over L of V[b,:,h,d]
// ---------------------------------------------------------------------------
__global__ __launch_bounds__(256) void vmean_kernel(const float* __restrict__ V,
                                                    float* __restrict__ vmean) {
  int bh = blockIdx.x;              // 0..127
  int b = bh >> 3, h = bh & 7;
  int t = threadIdx.x;
  int d = t & 63;                   // 0..63
  int lc = t >> 6;                  // 0..3
  float acc = 0.f;
  for (int l = lc; l < L_; l += 4)
    acc += V[(((size_t)b * L_ + l) * H_ + h) * D_ + d];
  __shared__ float red[256];
  red[t] = acc;
  __syncthreads();
  if (t < 128) red[t] += red[t + 128];
  __syncthreads();
  if (t < 64) vmean[(size_t)bh * 64 + t] = (red[t] + red[t + 64]) * (1.0f / L_);
}

// ---------------------------------------------------------------------------
// Kernel 1b: out[b,l,h,d] = vmean[b,h,d]  (default context, full 134 MB write)
// ---------------------------------------------------------------------------
__global__ __launch_bounds__(256) void fill_kernel(const float* __restrict__ vmean,
                                                   float* __restrict__ out) {
  size_t idx = (size_t)blockIdx.x * 256 + threadIdx.x;   // d fastest, then h, l, b
  int d = (int)(idx & 63);
  size_t r = idx >> 6;
  int h = (int)(r & 7);
  size_t b = r >> 15;
  out[idx] = vmean[((b * H_) + h) * 64 + d];
}

// ---------------------------------------------------------------------------
// Kernel 2: sparsity[bh*L + q] = max_s(q.k_s) - sum_s(q.k_s)/L
// One wave per query; lanes own sampled keys; single cross-lane reduction.
// ---------------------------------------------------------------------------
__global__ __launch_bounds__(256) void sparsity_kernel(const float* __restrict__ Q,
                                                       const float* __restrict__ K,
                                                       const int* __restrict__ samp,
                                                       float* __restrict__ spars) {
  int wave = threadIdx.x >> 5;
  int lane = threadIdx.x & 31;
  size_t gq = (size_t)blockIdx.x * 8 + wave;    // global query id = bh*L + q
  int q = (int)(gq & (L_ - 1));
  int bh = (int)(gq >> 12);
  int b = bh >> 3, h = bh & 7;
  const float* qrow = Q + (((size_t)b * L_ + q) * H_ + h) * D_;
  v2f qr[32];                                   // full Q row in registers
#pragma unroll
  for (int dd = 0; dd < 32; ++dd)
    qr[dd] = *reinterpret_cast<const v2f*>(qrow + dd * 2);

  float mx = NEG_INF_, sm = 0.f;
#pragma unroll
  for (int round = 0; round < 2; ++round) {
    int s = round * 32 + lane;
    bool valid = (s < SK_);
    int kidx = samp[q * SK_ + (valid ? s : 0)];
    const float* krow = K + (((size_t)b * L_ + kidx) * H_ + h) * D_;
    float dot = 0.f;
#pragma unroll
    for (int dd = 0; dd < 32; ++dd) {
      v2f kv = *reinterpret_cast<const v2f*>(krow + dd * 2);
      dot += qr[dd].x * kv.x + qr[dd].y * kv.y;
    }
    mx = fmaxf(mx, valid ? dot : NEG_INF_);
    sm += valid ? dot : 0.f;
  }
  // one wave-wide max+sum reduction (5 steps each)
#pragma unroll
  for (int msk = 1; msk < 32; msk <<= 1) {
    mx = fmaxf(mx, __shfl_xor(mx, msk, 32));
    sm += __shfl_xor(sm, msk, 32);
  }
  if (lane == 0) spars[gq] = mx - sm * (1.0f / L_);
}

// ---------------------------------------------------------------------------
// Kernel 3: per (b,h) iterative top-45 (argmax + mask, values live in LDS)
// ---------------------------------------------------------------------------
__global__ __launch_bounds__(256) void topk_kernel(const float* __restrict__ spars,
                                                   int* __restrict__ mtop) {
  int bh = blockIdx.x;
  int t = threadIdx.x;
  __shared__ float vals[L_];
  __shared__ float rv[256];
  __shared__ int ri[256];
  __shared__ int chosen;
  for (int i = t; i < L_; i += 256) vals[i] = spars[(size_t)bh * L_ + i];
  __syncthreads();
  for (int it = 0; it < NTOP_; ++it) {
    float bv = NEG_INF_;
    int bi = 0;
    for (int i = t; i < L_; i += 256) {
      float v = vals[i];
      if (v > bv) { bv = v; bi = i; }
    }
    rv[t] = bv; ri[t] = bi;
    __syncthreads();
    for (int off = 128; off > 0; off >>= 1) {
      if (t < off && rv[t + off] > rv[t]) { rv[t] = rv[t + off]; ri[t] = ri[t + off]; }
      __syncthreads();
    }
    if (t == 0) { chosen = ri[0]; mtop[bh * 48 + it] = ri[0]; }
    __syncthreads();
    if (t == 0) vals[chosen] = NEG_INF_;
    __syncthreads();
  }
  if (t < 3) mtop[bh * 48 + NTOP_ + t] = 0;   // pad rows 45..47
}

// ---------------------------------------------------------------------------
// Kernel 4: flash attention over selected queries, key-split partials.
// One wave handles a 16-query tile; 64 keys (4 S-tiles) per softmax update.
// ---------------------------------------------------------------------------
__global__ __launch_bounds__(32) void flash_kernel(const float* __restrict__ Q,
                                                   const float* __restrict__ K,
                                                   const float* __restrict__ V,
                                                   const int* __restrict__ mtop,
                                                   float* __restrict__ part,
                                                   int nchunks) {
  int chunk = blockIdx.x;
  int mtile = blockIdx.y;
  int bh = blockIdx.z;
  int b = bh >> 3, h = bh & 7;
  int lane = threadIdx.x;
  int half = lane >> 4;           // 0: K+0..1, 1: K+2..3
  int l16 = lane & 15;

  int mrow = mtile * 16 + l16;
  int q = (mrow < NTOP_) ? mtop[bh * 48 + mrow] : 0;

  // Preload Q A-fragments for all 16 k-steps (lane's half of its query row).
  const float* qrow = Q + (((size_t)b * L_ + q) * H_ + h) * D_;
  v2f aq[16];
#pragma unroll
  for (int kk = 0; kk < 16; ++kk)
    aq[kk] = *reinterpret_cast<const v2f*>(qrow + kk * 4 + 2 * half);

  v8f o[4];
#pragma unroll
  for (int dt = 0; dt < 4; ++dt) o[dt] = {};
  float mrun[8], lrun[8];
#pragma unroll
  for (int r = 0; r < 8; ++r) { mrun[r] = NEG_INF_; lrun[r] = 0.f; }

  __shared__ float plds[16 * PSTRIDE_];   // padded 16x64 P tile

  int keys_per_chunk = L_ / nchunks;
  int n0 = chunk * keys_per_chunk;

  for (int nb = 0; nb < keys_per_chunk; nb += 64) {
    int nbase = n0 + nb;

    // ---- S tiles: c[nt] = Q * K^T for 4 consecutive 16-key tiles ----
    v8f c[4];
    const float* kp[4];
#pragma unroll
    for (int nt = 0; nt < 4; ++nt) {
      c[nt] = {};
      kp[nt] = K + (((size_t)b * L_ + (nbase + nt * 16 + l16)) * H_ + h) * D_ + 2 * half;
    }
#pragma unroll
    for (int kk = 0; kk < 16; ++kk) {
#pragma unroll
      for (int nt = 0; nt < 4; ++nt) {       // 4 independent WMMA chains
        v2f bk = *reinterpret_cast<const v2f*>(kp[nt] + kk * 4);
        c[nt] = __builtin_amdgcn_wmma_f32_16x16x4_f32(false, aq[kk], false, bk,
                                                      (short)0, c[nt], false, false);
      }
    }

    // ---- online softmax over 64 keys at once ----
#pragma unroll
    for (int r = 0; r < 8; ++r) {
      float s0 = c[0][r] * SCALE_, s1 = c[1][r] * SCALE_;
      float s2 = c[2][r] * SCALE_, s3 = c[3][r] * SCALE_;
      float mx = fmaxf(fmaxf(s0, s1), fmaxf(s2, s3));
      mx = fmaxf(mx, __shfl_xor(mx, 1, 32));
      mx = fmaxf(mx, __shfl_xor(mx, 2, 32));
      mx = fmaxf(mx, __shfl_xor(mx, 4, 32));
      mx = fmaxf(mx, __shfl_xor(mx, 8, 32));
      float mnew = fmaxf(mrun[r], mx);
      float p0 = __expf(s0 - mnew), p1 = __expf(s1 - mnew);
      float p2 = __expf(s2 - mnew), p3 = __expf(s3 - mnew);
      float corr = __expf(mrun[r] - mnew);
      float ps = (p0 + p1) + (p2 + p3);
      ps += __shfl_xor(ps, 1, 32);
      ps += __shfl_xor(ps, 2, 32);
      ps += __shfl_xor(ps, 4, 32);
      ps += __shfl_xor(ps, 8, 32);
      lrun[r] = lrun[r] * corr + ps;
      mrun[r] = mnew;
#pragma unroll
      for (int dt = 0; dt < 4; ++dt) o[dt][r] *= corr;
      int rowoff = (r + 8 * half) * PSTRIDE_ + l16;
      plds[rowoff]      = p0;
      plds[rowoff + 16] = p1;
      plds[rowoff + 32] = p2;
      plds[rowoff + 48] = p3;
    }
    __syncthreads();   // single-wave group: S_NOP; LDS in-order within wave

    // ---- O += P * V (K=64 in 16 steps; 4 independent dt chains) ----
#pragma unroll
    for (int kk = 0; kk < 16; ++kk) {
      int kr = kk * 4 + 2 * half;
      v2f a = *reinterpret_cast<const v2f*>(&plds[l16 * PSTRIDE_ + kr]);
      const float* v0 = V + (((size_t)b * L_ + (nbase + kr)) * H_ + h) * D_;
      const float* v1 = v0 + (size_t)H_ * D_;
#pragma unroll
      for (int dt = 0; dt < 4; ++dt) {
        int d = dt * 16 + l16;
        v2f bv;
        bv.x = v0[d];
        bv.y = v1[d];
        o[dt] = __builtin_amdgcn_wmma_f32_16x16x4_f32(false, a, false, bv,
                                                      (short)0, o[dt], false, false);
      }
    }
    __syncthreads();
  }

  // ---- write partial (O, m, l) ----
  float* pb = part + ((size_t)(bh * MTILES_ + mtile) * nchunks + chunk) * PART_STRIDE_;
#pragma unroll
  for (int dt = 0; dt < 4; ++dt)
#pragma unroll
    for (int r = 0; r < 8; ++r)
      pb[(r + 8 * half) * 64 + dt * 16 + l16] = o[dt][r];
  if (l16 == 0) {
#pragma unroll
    for (int r = 0; r < 8; ++r) {
      pb[1024 + r + 8 * half] = mrun[r];
      pb[1040 + r + 8 * half] = lrun[r];
    }
  }
}

// ---------------------------------------------------------------------------
// Kernel 5: merge key-split partials, normalize, scatter rows into out
// ---------------------------------------------------------------------------
__global__ __launch_bounds__(256) void combine_kernel(const float* __restrict__ part,
                                                      const int* __restrict__ mtop,
                                                      float* __restrict__ out,
                                                      int nchunks) {
  int bm = blockIdx.x;                 // bh*3 + mtile
  int bh = bm / MTILES_, mtile = bm % MTILES_;
  int b = bh >> 3, h = bh & 7;
  const float* pb = part + (size_t)bm * nchunks * PART_STRIDE_;
  for (int e = threadIdx.x; e < 1024; e += 256) {
    int m = e >> 6, d = e & 63;
    int mrow = mtile * 16 + m;
    if (mrow >= NTOP_) continue;
    float M = NEG_INF_;
    for (int c = 0; c < nchunks; ++c)
      M = fmaxf(M, pb[c * PART_STRIDE_ + 1024 + m]);
    float lt = 0.f, oc = 0.f;
    for (int c = 0; c < nchunks; ++c) {
      float w = __expf(pb[c * PART_STRIDE_ + 1024 + m] - M);
      lt += pb[c * PART_STRIDE_ + 1040 + m] * w;
      oc += pb[c * PART_STRIDE_ + m * 64 + d] * w;
    }
    int qpos = mtop[bh * 48 + mrow];
    out[(((size_t)b * L_ + qpos) * H_ + h) * D_ + d] = oc / lt;
  }
}

// ---------------------------------------------------------------------------
extern "C" void kernel_launch(void* const* d_in, const int* in_sizes, int n_in,
                              void* d_out, int out_size, void* d_ws, size_t ws_size,
                              hipStream_t stream) {
  const float* Q = (const float*)d_in[0];
  const float* K = (const float*)d_in[1];
  const float* V = (const float*)d_in[2];
  const int* samp = (const int*)d_in[3];
  float* out = (float*)d_out;

  // workspace layout (float units)
  float* vmean = (float*)d_ws;                         // 8192
  float* spars = vmean + (size_t)B_ * H_ * D_;         // 524288
  int* mtop = (int*)(spars + (size_t)B_ * H_ * L_);    // 128*48 ints
  float* part = (float*)(mtop + 128 * 48);

  // pick largest key-split that fits in workspace (flash-decoding)
  size_t base_floats = (size_t)B_ * H_ * D_ + (size_t)B_ * H_ * L_ + 128 * 48;
  int nchunks = 8;
  while (nchunks > 1 &&
         (base_floats + (size_t)128 * MTILES_ * nchunks * PART_STRIDE_) * 4 > ws_size)
    nchunks >>= 1;

  vmean_kernel<<<B_ * H_, 256, 0, stream>>>(V, vmean);
  fill_kernel<<<(B_ * L_ * H_ * D_) / 256, 256, 0, stream>>>(vmean, out);
  sparsity_kernel<<<(B_ * H_ * L_) / 8, 256, 0, stream>>>(Q, K, samp, spars);
  topk_kernel<<<B_ * H_, 256, 0, stream>>>(spars, mtop);
  flash_kernel<<<dim3(nchunks, MTILES_, B_ * H_), 32, 0, stream>>>(Q, K, V, mtop, part,
                                                                   nchunks);
  combine_kernel<<<B_ * H_ * MTILES_, 256, 0, stream>>>(part, mtop, out, nchunks);
}